// ClasswiseECELoss_47012712022077
// MI455X (gfx1250) — compile-verified
//
#include <hip/hip_runtime.h>
#include <hip/hip_bf16.h>

#define NROWS 50000
#define NCLS  1000
#define NB    15
#define NSEG  (NCLS * NB)          // 15000 segments
#define TB    256                  // 8 waves (wave32)
#define GRID_MAIN 512
#define ACTIVE_T  (NCLS / 4)       // 250 threads cover 1000 floats, 4 each (b128)
#define DEPTH 4                    // async pipeline depth (rows in flight/block)
#define FIN_BLOCKS 8

// ---------------------------------------------------------------------------
// CDNA5 async global->LDS staging (ASYNCcnt path, ISA ch.08 / 15.18.3).
// Inline asm: portable across ROCm 7.2 and amdgpu-toolchain builtin arities.
// Operands: vdst(LDS byte addr), vaddr(64b global), off.  th:TH_LOAD_NT —
// the 200MB stream is touched once, keep it out of the 192MB L2.
// Low 32 bits of a flat pointer into __shared__ are the wave-relative LDS
// byte offset (ISA 10.2 aperture rules) == what VDST expects.
// ---------------------------------------------------------------------------
__device__ __forceinline__ void async_copy_b128(const float* gptr, const float* lds_ptr) {
    unsigned long long ga = (unsigned long long)(const void*)gptr;
    unsigned lo = (unsigned)(unsigned long long)(const void*)lds_ptr;
    asm volatile("global_load_async_to_lds_b128 %0, %1, off th:TH_LOAD_NT"
                 :: "v"(lo), "v"(ga) : "memory");
}

__device__ __forceinline__ int bin_of(float p) {
    int b = (int)__builtin_ceilf(p * (float)NB) - 1;
    b = b < 0 ? 0 : b;
    b = b > (NB - 1) ? (NB - 1) : b;
    return b;
}

// ---------------------------------------------------------------------------
// Zero workspace accumulators + output (harness poisons with 0xAA).
// ---------------------------------------------------------------------------
__global__ void __launch_bounds__(TB) ece_zero_kernel(float* ws, int n, float* out) {
    int i = blockIdx.x * TB + threadIdx.x;
    if (i < n) ws[i] = 0.0f;
    if (i == 0) out[0] = 0.0f;
}

// ---------------------------------------------------------------------------
// Main streaming histogram. Each block privatizes the FULL (class,bin)
// histogram in LDS (120 KB — CDNA5's 320 KB/WGP makes this possible) and
// streams rows through a DEPTH-deep async-to-LDS pipeline. Each wave stages
// and consumes only its own lanes' bytes: no cross-wave sync in the loop.
// ---------------------------------------------------------------------------
__global__ void __launch_bounds__(TB)
ece_hist_kernel(const float* __restrict__ sm,
                float* __restrict__ g_conf,
                float* __restrict__ g_cnt) {
    __shared__ float hconf[NSEG];                        // 60 KB
    __shared__ float hcnt [NSEG];                        // 60 KB
    __shared__ __align__(16) float stage[DEPTH][1024];   // 4 x 4 KB row buffers

    const int t = threadIdx.x;
    for (int i = t; i < NSEG; i += TB) { hconf[i] = 0.0f; hcnt[i] = 0.0f; }
    __syncthreads();

    const bool active = (t < ACTIVE_T);                  // lanes owning 4 classes
    const int  c0 = t * 4;

    const int row0 = blockIdx.x;
    // prologue: fill DEPTH-1 buffers
    #pragma unroll
    for (int k = 0; k < DEPTH - 1; ++k) {
        const int r = row0 + k * GRID_MAIN;
        if (r < NROWS && active)
            async_copy_b128(sm + (unsigned long long)r * NCLS + c0, &stage[k][c0]);
    }

    int i = 0;
    for (int row = row0; row < NROWS; row += GRID_MAIN, ++i) {
        const int r1 = row + 1 * GRID_MAIN;              // all block-uniform
        const int r2 = row + 2 * GRID_MAIN;
        const int r3 = row + 3 * GRID_MAIN;
        if (r3 < NROWS && active)
            async_copy_b128(sm + (unsigned long long)r3 * NCLS + c0,
                            &stage[(i + 3) & (DEPTH - 1)][c0]);

        // in-order completion: wait until this row's load (oldest) is done.
        // outstanding-younger = #{r1,r2,r3 issued}; conditions nest.
        if      (r3 < NROWS) asm volatile("s_wait_asynccnt 0x3" ::: "memory");
        else if (r2 < NROWS) asm volatile("s_wait_asynccnt 0x2" ::: "memory");
        else if (r1 < NROWS) asm volatile("s_wait_asynccnt 0x1" ::: "memory");
        else                 asm volatile("s_wait_asynccnt 0x0" ::: "memory");

        if (active) {
            float4 v = *(const float4*)&stage[i & (DEPTH - 1)][c0];  // ds_load_b128
            float p[4] = { v.x, v.y, v.z, v.w };
            #pragma unroll
            for (int j = 0; j < 4; ++j) {
                const int b = bin_of(p[j]);
                const int s = (c0 + j) * NB + b;
                atomicAdd(&hconf[s], p[j]);              // ds_add_f32
                atomicAdd(&hcnt [s], 1.0f);              // ds_add_f32
            }
        }
    }

    __syncthreads();
    // flush: skip untouched (class,bin) cells to cut global atomic traffic
    for (int k = t; k < NSEG; k += TB) {
        const float c = hcnt[k];
        if (c != 0.0f) {
            atomicAdd(&g_cnt [k], c);
            atomicAdd(&g_conf[k], hconf[k]);
        }
    }
}

// ---------------------------------------------------------------------------
// Correct-term: only (n, label[n]) contributes — 50K elements, not 50M.
// ---------------------------------------------------------------------------
__global__ void __launch_bounds__(TB)
ece_corr_kernel(const float* __restrict__ sm,
                const int* __restrict__ labels,
                float* __restrict__ g_corr) {
    const int n = blockIdx.x * TB + threadIdx.x;
    if (n >= NROWS) return;
    const int c = labels[n];
    const float p = sm[(unsigned long long)n * NCLS + c];
    atomicAdd(&g_corr[c * NB + bin_of(p)], 1.0f);
}

// ---------------------------------------------------------------------------
// Finalize: sum_{c,b} (cnt/N)*|conf-corr|/cnt, scaled by 1/C.
// Split over FIN_BLOCKS blocks; partials merged with one global atomic each.
// ---------------------------------------------------------------------------
__global__ void __launch_bounds__(TB)
ece_final_kernel(const float* __restrict__ g_conf,
                 const float* __restrict__ g_cnt,
                 const float* __restrict__ g_corr,
                 float* __restrict__ out) {
    __shared__ float red[TB];
    float s = 0.0f;
    for (int i = blockIdx.x * TB + threadIdx.x; i < NSEG; i += FIN_BLOCKS * TB) {
        const float c = g_cnt[i];
        if (c > 0.0f) {
            const float gap = __builtin_fabsf(g_conf[i] - g_corr[i]) / c;
            s += c * (1.0f / (float)NROWS) * gap;
        }
    }
    red[threadIdx.x] = s;
    __syncthreads();
    for (int off = TB / 2; off > 0; off >>= 1) {
        if (threadIdx.x < off) red[threadIdx.x] += red[threadIdx.x + off];
        __syncthreads();
    }
    if (threadIdx.x == 0) atomicAdd(out, red[0] * (1.0f / (float)NCLS));
}

// ---------------------------------------------------------------------------
extern "C" void kernel_launch(void* const* d_in, const int* in_sizes, int n_in,
                              void* d_out, int out_size, void* d_ws, size_t ws_size,
                              hipStream_t stream) {
    (void)in_sizes; (void)n_in; (void)out_size; (void)ws_size;
    const float* sm     = (const float*)d_in[0];
    const int*   labels = (const int*)d_in[1];
    float* out = (float*)d_out;

    float* g_conf = (float*)d_ws;             // [NSEG]
    float* g_cnt  = g_conf + NSEG;            // [NSEG]
    float* g_corr = g_cnt + NSEG;             // [NSEG]

    const int nzero = 3 * NSEG;
    ece_zero_kernel<<<(nzero + TB - 1) / TB, TB, 0, stream>>>(g_conf, nzero, out);
    ece_hist_kernel<<<GRID_MAIN, TB, 0, stream>>>(sm, g_conf, g_cnt);
    ece_corr_kernel<<<(NROWS + TB - 1) / TB, TB, 0, stream>>>(sm, labels, g_corr);
    ece_final_kernel<<<FIN_BLOCKS, TB, 0, stream>>>(g_conf, g_cnt, g_corr, out);
}